// ERFNet_33225867001923
// MI455X (gfx1250) — compile-verified
//
#include <hip/hip_runtime.h>

// ---------------- geometry ----------------
#define IMGN   16        // batch
#define IMG    1024      // H = W
#define TILE   64        // output tile per workgroup
#define HXL    24        // left x halo (padded from 20 to 24 for 16B alignment)
#define PW     112       // patch width  = 24 + 64 + 24
#define PH     104       // patch height = 20 + 64 + 20
#define NL     20        // layers
#define NTHR   256       // 8 waves (wave32)

typedef __attribute__((address_space(1))) void as1_void;
typedef __attribute__((address_space(3))) void as3_void;

typedef int v4i __attribute__((ext_vector_type(4)));
typedef __attribute__((address_space(1))) v4i gas_v4i;   // int4 in global addrspace
typedef __attribute__((address_space(3))) v4i las_v4i;   // int4 in LDS addrspace

#if __has_builtin(__builtin_amdgcn_global_load_async_to_lds_b128)
#define HAVE_ASYNC_LD 1
#endif
#if __has_builtin(__builtin_amdgcn_global_store_async_from_lds_b128)
#define HAVE_ASYNC_ST 1
#endif

__device__ __forceinline__ void async_ld16(const float* g, float* l) {
#ifdef HAVE_ASYNC_LD
    // global -> LDS, 16 bytes per lane, ASYNCcnt-tracked DMA (no VGPR round-trip)
    __builtin_amdgcn_global_load_async_to_lds_b128(
        (gas_v4i*)(as1_void*)g, (las_v4i*)(as3_void*)l, /*offset*/0, /*cpol*/0);
#else
    *(float4*)l = *(const float4*)g;   // fallback: load + ds_store
#endif
}

__device__ __forceinline__ void async_st16(float* g, const float* l) {
#ifdef HAVE_ASYNC_ST
    __builtin_amdgcn_global_store_async_from_lds_b128(
        (gas_v4i*)(as1_void*)g, (las_v4i*)(as3_void*)l, /*offset*/0, /*cpol*/0);
#else
    *(float4*)g = *(const float4*)l;
#endif
}

__device__ __forceinline__ void wait_async0() {
#if __has_builtin(__builtin_amdgcn_s_wait_asynccnt)
    __builtin_amdgcn_s_wait_asynccnt(0);
#else
    asm volatile("s_wait_asynccnt 0" ::: "memory");
#endif
}

extern "C" __global__ void __launch_bounds__(NTHR)
spike_conv20_fused(const float* __restrict__ xin,
                   const float* __restrict__ Wg,   // [20*9]
                   const float* __restrict__ bg,   // [20]
                   float* __restrict__ out)
{
    extern __shared__ __align__(16) float smem[];
    float* bufA = smem;                    // PH*PW
    float* bufB = smem + PH * PW;          // PH*PW
    float* wb   = smem + 2 * PH * PW;      // 180 weights + 20 biases

    const int tid = threadIdx.x;
    const int gx0 = blockIdx.x * TILE;
    const int gy0 = blockIdx.y * TILE;
    const int img = blockIdx.z;
    const int x0  = gx0 - HXL;             // global x of patch col 0 (mult of 4)
    const int y0  = gy0 - 20;              // global y of patch row 0

    const float* src_img = xin + (size_t)img * (IMG * (size_t)IMG);

    // stage weights + biases into LDS once
    if (tid < NL * 9 + NL)
        wb[tid] = (tid < NL * 9) ? Wg[tid] : bg[tid - NL * 9];

    // ---- async DMA of the 104x112 patch into LDS (16B vectors) ----
    // every 4-wide vector is fully inside or fully outside the image
    for (int v = tid; v < (PW / 4) * PH; v += NTHR) {
        const int vy = v / (PW / 4);
        const int vx = v - vy * (PW / 4);
        const int gx = x0 + 4 * vx;
        const int gy = y0 + vy;
        float* l = bufA + vy * PW + 4 * vx;
        if ((unsigned)gx < (unsigned)IMG && (unsigned)gy < (unsigned)IMG) {
            async_ld16(src_img + (size_t)gy * IMG + gx, l);
        } else {
            float4 z = {0.f, 0.f, 0.f, 0.f};
            *(float4*)l = z;               // zero padding
        }
    }
    wait_async0();
    __syncthreads();

    // ---- 20 fused conv+quant layers, ping-pong in LDS ----
    // Each thread owns one column (tx) and slides a 3x3 register window down
    // a contiguous row chunk (tz selects chunk): 3 LDS loads + 1 store per
    // pixel instead of 9 + 1.
    float* src = bufA;
    float* dst = bufB;
    const int tx = tid & 127;              // 128 columns covered (max 110 needed)
    const int tz = tid >> 7;               // 2 row chunks

    for (int layer = 0; layer < NL; ++layer) {
        const float w0 = wb[layer * 9 + 0], w1 = wb[layer * 9 + 1], w2 = wb[layer * 9 + 2];
        const float w3 = wb[layer * 9 + 3], w4 = wb[layer * 9 + 4], w5 = wb[layer * 9 + 5];
        const float w6 = wb[layer * 9 + 6], w7 = wb[layer * 9 + 7], w8 = wb[layer * 9 + 8];
        const float bias = wb[NL * 9 + layer];

        const int lo = layer + 1;          // valid region [lo, PW-lo) x [lo, PH-lo)
        const int hi = PH - lo;
        const int x  = lo + tx;
        if (x < PW - lo) {
            const int  gx   = x0 + x;
            const bool gxin = (unsigned)gx < (unsigned)IMG;

            const int half = ((hi - lo) + 1) >> 1;
            const int ys   = lo + tz * half;
            const int ye   = (ys + half < hi) ? (ys + half) : hi;

            // init window: rows ys-1 (a) and ys (b); both inside src's valid rows
            const float* pa = src + (ys - 1) * PW + x;
            float a0 = pa[-1], a1 = pa[0], a2 = pa[1];
            const float* pb = pa + PW;
            float b0 = pb[-1], b1 = pb[0], b2 = pb[1];

            for (int y = ys; y < ye; ++y) {
                const float* pc = src + (y + 1) * PW + x;
                const float c0 = pc[-1], c1 = pc[0], c2 = pc[1];
                float acc = bias;
                acc = fmaf(a0, w0, acc);
                acc = fmaf(a1, w1, acc);
                acc = fmaf(a2, w2, acc);
                acc = fmaf(b0, w3, acc);
                acc = fmaf(b1, w4, acc);
                acc = fmaf(b2, w5, acc);
                acc = fmaf(c0, w6, acc);
                acc = fmaf(c1, w7, acc);
                acc = fmaf(c2, w8, acc);
                // MultiSpike: round(clip(x, 0, 16)); rintf == round-half-even
                acc = rintf(fminf(fmaxf(acc, 0.0f), 16.0f));
                const int gy = y0 + y;
                const bool in = gxin && ((unsigned)gy < (unsigned)IMG);
                dst[y * PW + x] = in ? acc : 0.0f;   // keep zero-padding invariant
                a0 = b0; a1 = b1; a2 = b2;           // slide window down one row
                b0 = c0; b1 = c1; b2 = c2;
            }
        }
        __syncthreads();
        float* t = src; src = dst; dst = t;          // 20 swaps -> final data in src
    }

    // ---- async store of the 64x64 interior back to global ----
    float* dst_img = out + (size_t)img * (IMG * (size_t)IMG);
    for (int v = tid; v < (TILE / 4) * TILE; v += NTHR) {
        const int vy = v >> 4;             // 16 vectors per row
        const int vx = v & 15;
        const float* l = src + (vy + 20) * PW + (HXL + 4 * vx);
        async_st16(dst_img + (size_t)(gy0 + vy) * IMG + (gx0 + 4 * vx), l);
    }
    wait_async0();   // S_ENDPGM also does an implicit wait-idle
}

extern "C" void kernel_launch(void* const* d_in, const int* in_sizes, int n_in,
                              void* d_out, int out_size, void* d_ws, size_t ws_size,
                              hipStream_t stream) {
    (void)in_sizes; (void)n_in; (void)out_size; (void)d_ws; (void)ws_size;
    const float* x = (const float*)d_in[0];   // [16,1,1024,1024]
    const float* W = (const float*)d_in[1];   // [20,1,1,3,3]
    const float* b = (const float*)d_in[2];   // [20,1]
    float* out = (float*)d_out;

    dim3 grid(IMG / TILE, IMG / TILE, IMGN);  // 16 x 16 x 16 tiles
    dim3 block(NTHR);
    size_t shmem = (size_t)(2 * PW * PH + NL * 9 + NL) * sizeof(float); // ~92 KB
    spike_conv20_fused<<<grid, block, shmem, stream>>>(x, W, b, out);
}